// FusionMamba_45071386805024
// MI455X (gfx1250) — compile-verified
//
#include <hip/hip_runtime.h>

// ---------------- problem constants (from reference) ----------------
#define LSEQ 2304   // H*W = 48*48
#define CDIM 64     // DIM
#define CDIN 128    // 2*DIM
#define CDS  30     // SSM state size
#define CKC  4      // depthwise conv kernel

typedef __attribute__((ext_vector_type(2))) float v2f;
typedef __attribute__((ext_vector_type(8))) float v8f;

__device__ __forceinline__ float silu_f(float v) { return v * (1.0f / (1.0f + __expf(-v))); }
__device__ __forceinline__ float softplus_f(float v) { return (v > 20.0f) ? v : log1pf(__expf(v)); }

__device__ __forceinline__ v8f wmma4(v2f a, v2f b, v8f c) {
  // V_WMMA_F32_16X16X4_F32 : D(16x16 f32) = A(16x4 f32) * B(4x16 f32) + C
  return __builtin_amdgcn_wmma_f32_16x16x4_f32(false, a, false, b, (short)0, c, false, false);
}

// ---------------- layout shuffles ----------------
// img (DIM, L) -> seq (L, DIM)
__global__ void toseq_kernel(const float* __restrict__ img, float* __restrict__ seq) {
  int i = blockIdx.x * 256 + threadIdx.x;       // i = d*L + t  (coalesced read)
  if (i >= CDIM * LSEQ) return;
  int d = i / LSEQ, t = i - d * LSEQ;
  seq[t * CDIM + d] = img[i];
}
// seq (L, DIM) -> img (DIM, L)
__global__ void toimg_kernel(const float* __restrict__ seq, float* __restrict__ img) {
  int i = blockIdx.x * 256 + threadIdx.x;       // i = t*DIM + d (coalesced read)
  if (i >= CDIM * LSEQ) return;
  int t = i / CDIM, d = i - t * CDIM;
  img[d * LSEQ + t] = seq[i];
}

// ---------------- stage 1: LN + x = a@Wx^T, z = e@Wz^T ----------------
// grid = L/16, block = 256 (8 waves). Wx/Wz are (128,64).
__global__ void pre_kernel(const float* __restrict__ x0, const float* __restrict__ x1,
                           const float* __restrict__ ln0g, const float* __restrict__ ln0b,
                           const float* __restrict__ ln1g, const float* __restrict__ ln1b,
                           const float* __restrict__ Wx, const float* __restrict__ Wz,
                           float* __restrict__ xout, float* __restrict__ zout) {
  __shared__ float a_t[16][CDIM + 4];
  __shared__ float e_t[16][CDIM + 4];
  const int t0 = blockIdx.x * 16;
  const int tid = threadIdx.x;

  if (tid < 32) {  // layernorm of 16 tokens of x0 (tid<16) and x1 (tid>=16)
    int tok = tid & 15;
    const float* src = (tid < 16) ? x0 : x1;
    const float* g   = (tid < 16) ? ln0g : ln1g;
    const float* b   = (tid < 16) ? ln0b : ln1b;
    float* dst = (tid < 16) ? &a_t[tok][0] : &e_t[tok][0];
    const float* row = src + (size_t)(t0 + tok) * CDIM;
    float m = 0.f;
    #pragma unroll 8
    for (int k = 0; k < CDIM; ++k) m += row[k];
    m *= (1.0f / CDIM);
    float v = 0.f;
    #pragma unroll 8
    for (int k = 0; k < CDIM; ++k) { float d = row[k] - m; v += d * d; }
    float r = rsqrtf(v * (1.0f / CDIM) + 1e-5f);
    #pragma unroll 8
    for (int k = 0; k < CDIM; ++k) dst[k] = (row[k] - m) * r * g[k] + b[k];
  }
  __syncthreads();

  const int wave = tid >> 5, lane = tid & 31, col = lane & 15, half = lane >> 4;
  for (int job = 0; job < 2; ++job) {
    const float* W = job ? Wz : Wx;
    float* out = job ? zout : xout;
    const int nb = wave * 16;
    v8f acc = {};
    #pragma unroll
    for (int k0 = 0; k0 < CDIM; k0 += 4) {
      v2f a, b;
      const float* arow = job ? &e_t[col][0] : &a_t[col][0];
      a.x = arow[k0 + half * 2 + 0];
      a.y = arow[k0 + half * 2 + 1];
      b.x = W[(nb + col) * CDIM + k0 + half * 2 + 0];   // B[k][n] = W[n][k]
      b.y = W[(nb + col) * CDIM + k0 + half * 2 + 1];
      acc = wmma4(a, b, acc);
    }
    #pragma unroll
    for (int r = 0; r < 8; ++r)
      out[(size_t)(t0 + half * 8 + r) * CDIN + nb + col] = acc[r];
  }
}

// ---------------- stage 2: causal dwconv+SiLU, proj, dt ----------------
// grid = L/16, block = 256 (8 waves). Wxp (64,128), Wdt (128,4).
__global__ void conv_proj_kernel(const float* __restrict__ x, int rev,
                                 const float* __restrict__ cw, const float* __restrict__ cb,
                                 const float* __restrict__ Wxp,
                                 const float* __restrict__ Wdt, const float* __restrict__ bdt,
                                 float* __restrict__ xc_out, float* __restrict__ dt_out,
                                 float* __restrict__ B_out, float* __restrict__ C_out) {
  __shared__ float xc_t[16][CDIN + 4];
  __shared__ float pj_t[16][64 + 4];
  const int t0 = blockIdx.x * 16;
  const int tid = threadIdx.x;

  for (int i = tid; i < 16 * CDIN; i += 256) {  // depthwise causal conv + silu
    int tok = i >> 7, d = i & 127;
    int t = t0 + tok;
    float acc = cb[d];
    #pragma unroll
    for (int k = 0; k < CKC; ++k) {
      int ts = t - (CKC - 1) + k;
      float xv = 0.f;
      if (ts >= 0) {
        int src = rev ? (LSEQ - 1 - ts) : ts;
        xv = x[(size_t)src * CDIN + d];
      }
      acc += cw[d * CKC + k] * xv;
    }
    acc = silu_f(acc);
    xc_t[tok][d] = acc;
    xc_out[(size_t)t * CDIN + d] = acc;
  }
  __syncthreads();

  const int wave = tid >> 5, lane = tid & 31, col = lane & 15, half = lane >> 4;
  if (wave < 4) {  // proj = xc(16x128) @ Wxp^T -> (16x64), one N-tile per wave
    const int nb = wave * 16;
    v8f acc = {};
    #pragma unroll
    for (int k0 = 0; k0 < CDIN; k0 += 4) {
      v2f a, b;
      a.x = xc_t[col][k0 + half * 2 + 0];
      a.y = xc_t[col][k0 + half * 2 + 1];
      b.x = Wxp[(nb + col) * CDIN + k0 + half * 2 + 0];
      b.y = Wxp[(nb + col) * CDIN + k0 + half * 2 + 1];
      acc = wmma4(a, b, acc);
    }
    #pragma unroll
    for (int r = 0; r < 8; ++r) pj_t[half * 8 + r][nb + col] = acc[r];
  }
  __syncthreads();

  {  // dt = softplus(proj[:, :4] @ Wdt^T + bdt): K=4 -> exactly one WMMA per tile
    const int nb = wave * 16;
    v2f a, b;
    a.x = pj_t[col][half * 2 + 0];
    a.y = pj_t[col][half * 2 + 1];
    b.x = Wdt[(nb + col) * 4 + half * 2 + 0];
    b.y = Wdt[(nb + col) * 4 + half * 2 + 1];
    v8f acc = {};
    acc = wmma4(a, b, acc);
    #pragma unroll
    for (int r = 0; r < 8; ++r) {
      int t = t0 + half * 8 + r, d = nb + col;
      dt_out[(size_t)t * CDIN + d] = softplus_f(acc[r] + bdt[d]);
    }
  }

  for (int i = tid; i < 16 * 32; i += 256) {  // split out B (cols 4..33) and C (34..63)
    int tok = i >> 5, n = i & 31;
    float bv = (n < CDS) ? pj_t[tok][4 + n] : 0.f;
    float cv = (n < CDS) ? pj_t[tok][4 + CDS + n] : 0.f;
    B_out[(size_t)(t0 + tok) * 32 + n] = bv;
    C_out[(size_t)(t0 + tok) * 32 + n] = cv;
  }
}

// ---------------- stage 3: selective scan (sequential over L) ----------------
// grid = 2 (fwd/bwd), block = 128: thread d owns channel d; h[30],A[30] in VGPRs.
__global__ void scan_kernel(const float* __restrict__ xc_f, const float* __restrict__ dt_f,
                            const float* __restrict__ B_f, const float* __restrict__ C_f,
                            const float* __restrict__ Al_f, const float* __restrict__ Dv_f,
                            const float* __restrict__ xc_b, const float* __restrict__ dt_b,
                            const float* __restrict__ B_b, const float* __restrict__ C_b,
                            const float* __restrict__ Al_b, const float* __restrict__ Dv_b,
                            float* __restrict__ y_f, float* __restrict__ y_b) {
  const int dir = blockIdx.x;
  const float* xc = dir ? xc_b : xc_f;
  const float* dt = dir ? dt_b : dt_f;
  const float* Bm = dir ? B_b : B_f;
  const float* Cm = dir ? C_b : C_f;
  const float* Al = dir ? Al_b : Al_f;
  const float* Dv = dir ? Dv_b : Dv_f;
  float* yo = dir ? y_b : y_f;

  const int d = threadIdx.x;          // 0..127
  const int lane = threadIdx.x & 31;
  float A[CDS], h[CDS];
  #pragma unroll
  for (int n = 0; n < CDS; ++n) { A[n] = -__expf(Al[d * CDS + n]); h[n] = 0.f; }
  const float Dd = Dv[d];

  for (int t = 0; t < LSEQ; ++t) {
    float dtv = dt[(size_t)t * CDIN + d];
    float xv  = xc[(size_t)t * CDIN + d];
    float dtx = dtv * xv;
    float bl = (lane < CDS) ? Bm[(size_t)t * 32 + lane] : 0.f;
    float cl = (lane < CDS) ? Cm[(size_t)t * 32 + lane] : 0.f;
    float y = 0.f;
    #pragma unroll
    for (int n = 0; n < CDS; ++n) {
      float bn = __shfl(bl, n, 32);
      float cn = __shfl(cl, n, 32);
      float dA = __expf(dtv * A[n]);
      h[n] = dA * h[n] + dtx * bn;
      y += h[n] * cn;
    }
    y += xv * Dd;
    int tout = dir ? (LSEQ - 1 - t) : t;  // backward result stored pre-re-reversed
    yo[(size_t)tout * CDIN + d] = y;
  }
}

// ---------------- stage 4: gated combine + out-proj + skip ----------------
// grid = L/16, block = 128 (4 waves). Wout (64,128). dst may alias skip.
__global__ void combine_kernel(const float* __restrict__ yf, const float* __restrict__ yb,
                               const float* __restrict__ z, const float* __restrict__ Wout,
                               const float* __restrict__ skip, float* __restrict__ dst) {
  __shared__ float g_t[16][CDIN + 4];
  const int t0 = blockIdx.x * 16;
  const int tid = threadIdx.x;
  for (int i = tid; i < 16 * CDIN; i += 128) {
    int tok = i >> 7, dch = i & 127;
    size_t idx = (size_t)(t0 + tok) * CDIN + dch;
    g_t[tok][dch] = (yf[idx] + yb[idx]) * silu_f(z[idx]);
  }
  __syncthreads();
  const int wave = tid >> 5, lane = tid & 31, col = lane & 15, half = lane >> 4;
  const int nb = wave * 16;
  v8f acc = {};
  #pragma unroll
  for (int k0 = 0; k0 < CDIN; k0 += 4) {
    v2f a, b;
    a.x = g_t[col][k0 + half * 2 + 0];
    a.y = g_t[col][k0 + half * 2 + 1];
    b.x = Wout[(nb + col) * CDIN + k0 + half * 2 + 0];
    b.y = Wout[(nb + col) * CDIN + k0 + half * 2 + 1];
    acc = wmma4(a, b, acc);
  }
  #pragma unroll
  for (int r = 0; r < 8; ++r) {
    size_t idx = (size_t)(t0 + half * 8 + r) * CDIM + nb + col;
    dst[idx] = acc[r] + skip[idx];
  }
}

// ---------------- stage 5: final projection, writes image layout ----------------
// grid = L/16, block = 128. Wp (64,128); img is original input in (DIM,L) layout.
__global__ void final_kernel(const float* __restrict__ lbuf, const float* __restrict__ gbuf,
                             const float* __restrict__ Wp, const float* __restrict__ bp,
                             const float* __restrict__ img, float* __restrict__ outimg) {
  __shared__ float c_t[16][CDIN + 4];
  const int t0 = blockIdx.x * 16;
  const int tid = threadIdx.x;
  for (int i = tid; i < 16 * CDIN; i += 128) {
    int tok = i >> 7, c = i & 127;
    c_t[tok][c] = (c < CDIM) ? lbuf[(size_t)(t0 + tok) * CDIM + c]
                             : gbuf[(size_t)(t0 + tok) * CDIM + (c - CDIM)];
  }
  __syncthreads();
  const int wave = tid >> 5, lane = tid & 31, col = lane & 15, half = lane >> 4;
  const int nb = wave * 16;
  v8f acc = {};
  #pragma unroll
  for (int k0 = 0; k0 < CDIN; k0 += 4) {
    v2f a, b;
    a.x = c_t[col][k0 + half * 2 + 0];
    a.y = c_t[col][k0 + half * 2 + 1];
    b.x = Wp[(nb + col) * CDIN + k0 + half * 2 + 0];
    b.y = Wp[(nb + col) * CDIN + k0 + half * 2 + 1];
    acc = wmma4(a, b, acc);
  }
  #pragma unroll
  for (int r = 0; r < 8; ++r) {
    int t = t0 + half * 8 + r, m = nb + col;
    outimg[(size_t)m * LSEQ + t] = acc[r] + bp[m] + img[(size_t)m * LSEQ + t];
  }
}

// ---------------- host orchestration ----------------
extern "C" void kernel_launch(void* const* d_in, const int* in_sizes, int n_in,
                              void* d_out, int out_size, void* d_ws, size_t ws_size,
                              hipStream_t stream) {
  (void)in_sizes; (void)n_in; (void)out_size; (void)ws_size;
  const float* img1     = (const float*)d_in[0];
  const float* img2     = (const float*)d_in[1];
  const float* img1s    = (const float*)d_in[2];
  const float* img2s    = (const float*)d_in[3];
  const float* ln0g     = (const float*)d_in[4];
  const float* ln0b     = (const float*)d_in[5];
  const float* ln1g     = (const float*)d_in[6];
  const float* ln1b     = (const float*)d_in[7];
  const float* Wx       = (const float*)d_in[8];
  const float* Wz       = (const float*)d_in[9];
  const float* conv_w   = (const float*)d_in[10];
  const float* conv_b   = (const float*)d_in[11];
  const float* Wxp      = (const float*)d_in[12];
  const float* Wdt      = (const float*)d_in[13];
  const float* bdt      = (const float*)d_in[14];
  const float* A_log    = (const float*)d_in[15];
  const float* Dvec     = (const float*)d_in[16];
  const float* conv_w_b = (const float*)d_in[17];
  const float* conv_b_b = (const float*)d_in[18];
  const float* Wxp_b    = (const float*)d_in[19];
  const float* Wdt_b    = (const float*)d_in[20];
  const float* bdt_b    = (const float*)d_in[21];
  const float* A_log_b  = (const float*)d_in[22];
  const float* Dvec_b   = (const float*)d_in[23];
  const float* Wout     = (const float*)d_in[24];
  const float* Wp1      = (const float*)d_in[25];
  const float* bp1      = (const float*)d_in[26];
  const float* Wp2      = (const float*)d_in[27];
  const float* bp2      = (const float*)d_in[28];

  float* ws = (float*)d_ws;
  size_t off = 0;
  float* SEQ[8];                                   // s1,s2,ss1,ss2,l1,g1,l2,g2 (L,64)
  for (int i = 0; i < 8; ++i) { SEQ[i] = ws + off; off += (size_t)LSEQ * CDIM; }
  float* X   = ws + off; off += (size_t)LSEQ * CDIN;
  float* Z   = ws + off; off += (size_t)LSEQ * CDIN;
  float* XCF = ws + off; off += (size_t)LSEQ * CDIN;
  float* XCB = ws + off; off += (size_t)LSEQ * CDIN;
  float* DTF = ws + off; off += (size_t)LSEQ * CDIN;
  float* DTB = ws + off; off += (size_t)LSEQ * CDIN;
  float* YF  = ws + off; off += (size_t)LSEQ * CDIN;
  float* YB  = ws + off; off += (size_t)LSEQ * CDIN;
  float* BF  = ws + off; off += (size_t)LSEQ * 32;
  float* CF  = ws + off; off += (size_t)LSEQ * 32;
  float* BB  = ws + off; off += (size_t)LSEQ * 32;
  float* CB  = ws + off; off += (size_t)LSEQ * 32;

  const int TB = (CDIM * LSEQ + 255) / 256;
  toseq_kernel<<<TB, 256, 0, stream>>>(img1,  SEQ[0]);
  toseq_kernel<<<TB, 256, 0, stream>>>(img2,  SEQ[1]);
  toseq_kernel<<<TB, 256, 0, stream>>>(img1s, SEQ[2]);
  toseq_kernel<<<TB, 256, 0, stream>>>(img2s, SEQ[3]);

  // cross schedule: (x0, x1, block-idx == dst order) — matches reference data deps
  const int cx0[8]  = {0, 1, 2, 3, 0, 0, 1, 1};
  const int cx1[8]  = {2, 3, 0, 1, 1, 3, 0, 2};
  const int cdst[8] = {0, 1, 2, 3, 4, 5, 6, 7};
  const int NTB = LSEQ / 16;

  for (int c = 0; c < 8; ++c) {
    const int blk = c;
    const float* x0 = SEQ[cx0[c]];
    const float* x1 = SEQ[cx1[c]];
    float* dst = SEQ[cdst[c]];
    pre_kernel<<<NTB, 256, 0, stream>>>(
        x0, x1, ln0g + blk * CDIM, ln0b + blk * CDIM, ln1g + blk * CDIM, ln1b + blk * CDIM,
        Wx + (size_t)blk * CDIN * CDIM, Wz + (size_t)blk * CDIN * CDIM, X, Z);
    conv_proj_kernel<<<NTB, 256, 0, stream>>>(
        X, 0, conv_w + (size_t)blk * CDIN * CKC, conv_b + (size_t)blk * CDIN,
        Wxp + (size_t)blk * 64 * CDIN, Wdt + (size_t)blk * CDIN * 4, bdt + (size_t)blk * CDIN,
        XCF, DTF, BF, CF);
    conv_proj_kernel<<<NTB, 256, 0, stream>>>(
        X, 1, conv_w_b + (size_t)blk * CDIN * CKC, conv_b_b + (size_t)blk * CDIN,
        Wxp_b + (size_t)blk * 64 * CDIN, Wdt_b + (size_t)blk * CDIN * 4, bdt_b + (size_t)blk * CDIN,
        XCB, DTB, BB, CB);
    scan_kernel<<<2, 128, 0, stream>>>(
        XCF, DTF, BF, CF, A_log + (size_t)blk * CDIN * CDS, Dvec + (size_t)blk * CDIN,
        XCB, DTB, BB, CB, A_log_b + (size_t)blk * CDIN * CDS, Dvec_b + (size_t)blk * CDIN,
        YF, YB);
    combine_kernel<<<NTB, 128, 0, stream>>>(
        YF, YB, Z, Wout + (size_t)blk * CDIM * CDIN, x0, dst);
  }

  float* out = (float*)d_out;
  final_kernel<<<NTB, 128, 0, stream>>>(SEQ[4], SEQ[5], Wp1, bp1, img1, out);
  final_kernel<<<NTB, 128, 0, stream>>>(SEQ[6], SEQ[7], Wp2, bp2, img2, out + (size_t)CDIM * LSEQ);
  toimg_kernel<<<TB, 256, 0, stream>>>(SEQ[2], out + 2 * (size_t)CDIM * LSEQ);
  toimg_kernel<<<TB, 256, 0, stream>>>(SEQ[3], out + 3 * (size_t)CDIM * LSEQ);
}